// HAN_58497454571834
// MI455X (gfx1250) — compile-verified
//
#include <hip/hip_runtime.h>
#include <cmath>

#define NNODES 50000
#define NEDGES 800000
#define NMETA  2
#define DIN    256
#define DH     128
#define NHEADS 8
#define NCLS   16
#define NEG_SLOPE 0.2f

typedef float v2f __attribute__((ext_vector_type(2)));
typedef float v8f __attribute__((ext_vector_type(8)));

__device__ __forceinline__ float lrelu(float x) { return x > 0.f ? x : NEG_SLOPE * x; }
__device__ __forceinline__ float eluf(float x)  { return x > 0.f ? x : expm1f(x); }

// =====================================================================
// WMMA f32 GEMM (staged): Ncols == 128. Block = 8 waves = one 16-row tile,
// each wave one 16x16 col tile. A tile staged to LDS with async-to-LDS.
// MODE: 0 = raw, 1 = +bias then ELU
// =====================================================================
template <int MODE, int K>
__global__ void wmma_gemm_staged_kernel(const float* __restrict__ A,
                                        const float* __restrict__ B,
                                        const float* __restrict__ bias,
                                        float* __restrict__ C) {
  constexpr int LDA = K + 4;                    // pad: bank-conflict-free, 16B rows
  __shared__ float As[16 * LDA];
  const int rowbase = blockIdx.x << 4;
  const int t = threadIdx.x;

  // ---- stage A[16][K] -> LDS via async copy (16B chunks, coalesced) ----
  constexpr int CH = K / 4;                     // 16B chunks per row
#pragma unroll
  for (int c = t; c < 16 * CH; c += 256) {
    int row = c / CH;
    int kk = (c - row * CH) * 4;
    unsigned ldsByte = (unsigned)(uintptr_t)(&As[row * LDA + kk]);
    unsigned long long ga = (unsigned long long)(A + (size_t)(rowbase + row) * K + kk);
    asm volatile("global_load_async_to_lds_b128 %0, %1, off"
                 :: "v"(ldsByte), "v"(ga) : "memory");
  }
  asm volatile("s_wait_asynccnt 0" ::: "memory");
  __syncthreads();

  // ---- WMMA main loop ----
  const int lane = t & 31;
  const int ct = t >> 5;                        // col tile 0..7
  const int col = (ct << 4) + (lane & 15);
  const int khalf = (lane >> 4) << 1;           // lanes0-15 -> K{0,1}, lanes16-31 -> K{2,3}
  const float* __restrict__ Alds = &As[(lane & 15) * LDA];

  v8f acc = {};
#pragma unroll 4
  for (int k = 0; k < K; k += 4) {
    v2f a, b;
    a.x = Alds[k + khalf + 0];
    a.y = Alds[k + khalf + 1];
    b.x = B[(size_t)(k + khalf + 0) * DH + col];
    b.y = B[(size_t)(k + khalf + 1) * DH + col];
    acc = __builtin_amdgcn_wmma_f32_16x16x4_f32(false, a, false, b, (short)0, acc,
                                                false, false);
  }
  const int mbase = rowbase + ((lane >> 4) << 3);
  float bv = (MODE >= 1) ? bias[col] : 0.f;
#pragma unroll
  for (int r = 0; r < 8; ++r) {
    float v = acc[r];
    if (MODE >= 1) v = eluf(v + bv);
    C[(size_t)(mbase + r) * DH + col] = v;
  }
}

// =====================================================================
// WMMA f32 GEMM (direct): generic shapes, one wave per 16x16 tile.
// MODE: 2 = +bias (classifier)
// =====================================================================
template <int MODE>
__global__ void wmma_gemm_direct_kernel(const float* __restrict__ A,
                                        const float* __restrict__ B,
                                        const float* __restrict__ bias,
                                        float* __restrict__ C,
                                        int Mrows, int K, int Ncols) {
  const int wave = (int)((blockIdx.x * blockDim.x + threadIdx.x) >> 5);
  const int lane = threadIdx.x & 31;
  const int colTiles = Ncols >> 4;
  const int rowTiles = Mrows >> 4;
  if (wave >= rowTiles * colTiles) return;       // wave-uniform guard
  const int rt = wave / colTiles;
  const int ct = wave - rt * colTiles;

  const int arow  = (rt << 4) + (lane & 15);
  const int col   = (ct << 4) + (lane & 15);
  const int khalf = (lane >> 4) << 1;

  const float* __restrict__ Arow = A + (size_t)arow * K;
  v8f acc = {};
#pragma unroll 4
  for (int k = 0; k < K; k += 4) {
    v2f a, b;
    a.x = Arow[k + khalf + 0];
    a.y = Arow[k + khalf + 1];
    b.x = B[(size_t)(k + khalf + 0) * Ncols + col];
    b.y = B[(size_t)(k + khalf + 1) * Ncols + col];
    acc = __builtin_amdgcn_wmma_f32_16x16x4_f32(false, a, false, b, (short)0, acc,
                                                false, false);
  }
  const int mbase = (rt << 4) + ((lane >> 4) << 3);
  float bv = (MODE >= 1) ? bias[col] : 0.f;
#pragma unroll
  for (int r = 0; r < 8; ++r) {
    float v = acc[r] + bv;
    C[(size_t)(mbase + r) * Ncols + col] = v;
  }
}

// =====================================================================
// CSR build (dst-sorted src lists), built once per metapath
// =====================================================================
__global__ void csr_count_kernel(const int* __restrict__ dst, int* __restrict__ indeg) {
  int e = blockIdx.x * blockDim.x + threadIdx.x;
  if (e >= NEDGES) return;
  atomicAdd(&indeg[dst[e]], 1);
}

// single block, 1024 threads: exclusive scan indeg -> rowptr (and copy -> fill)
__global__ void csr_scan_kernel(const int* __restrict__ indeg, int* __restrict__ rowptr,
                                int* __restrict__ fill) {
  __shared__ int sh[1024];
  __shared__ int carrySh;
  const int t = threadIdx.x;
  if (t == 0) carrySh = 0;
  __syncthreads();
  for (int base = 0; base < NNODES; base += 1024) {
    int i = base + t;
    int v = (i < NNODES) ? indeg[i] : 0;
    sh[t] = v;
    __syncthreads();
#pragma unroll
    for (int off = 1; off < 1024; off <<= 1) {
      int add = (t >= off) ? sh[t - off] : 0;
      __syncthreads();
      sh[t] += add;
      __syncthreads();
    }
    int excl = sh[t] - v;
    int carry = carrySh;
    if (i < NNODES) { rowptr[i] = carry + excl; fill[i] = carry + excl; }
    __syncthreads();
    if (t == 1023) carrySh = carry + sh[1023];
    __syncthreads();
  }
  if (t == 0) rowptr[NNODES] = carrySh;
}

__global__ void csr_fill_kernel(const int* __restrict__ src, const int* __restrict__ dst,
                                int* __restrict__ fill, int* __restrict__ srcs) {
  int e = blockIdx.x * blockDim.x + threadIdx.x;
  if (e >= NEDGES) return;
  int d = dst[e];
  int pos = atomicAdd(&fill[d], 1);
  srcs[pos] = src[e];
}

// =====================================================================
// attention logit reductions
// =====================================================================
__global__ void elr8_kernel(const float* __restrict__ z, const float* __restrict__ al,
                            const float* __restrict__ ar, float* __restrict__ el,
                            float* __restrict__ er) {
  int i = blockIdx.x * blockDim.x + threadIdx.x;
  if (i >= NNODES * NHEADS) return;
  int n = i >> 3, h = i & 7;
  const float* zp = z + (size_t)n * DH + h * 16;
  const float* ap = al + h * 16;
  const float* rp = ar + h * 16;
  float s = 0.f, t = 0.f;
#pragma unroll
  for (int d = 0; d < 16; ++d) { float zv = zp[d]; s += zv * ap[d]; t += zv * rp[d]; }
  el[i] = s; er[i] = t;
}

__global__ void elr1_kernel(const float* __restrict__ z, const float* __restrict__ al,
                            const float* __restrict__ ar, float* __restrict__ el,
                            float* __restrict__ er) {
  int wave = (int)((blockIdx.x * blockDim.x + threadIdx.x) >> 5);
  int lane = threadIdx.x & 31;
  if (wave >= NNODES) return;
  const float* zp = z + (size_t)wave * DH;
  float s = 0.f, t = 0.f;
#pragma unroll
  for (int j = 0; j < 4; ++j) {
    int c = j * 32 + lane;
    float zv = zp[c];
    s += zv * al[c]; t += zv * ar[c];
  }
#pragma unroll
  for (int m = 16; m > 0; m >>= 1) { s += __shfl_xor(s, m, 32); t += __shfl_xor(t, m, 32); }
  if (lane == 0) { el[wave] = s; er[wave] = t; }
}

// =====================================================================
// Fused per-node GAT: edge-softmax (max/sum) + weighted gather + bias(+ELU).
// One wave per destination node; CSR gather, zero atomics.
// MODE: 1 = elu(acc + bias), 2 = acc + bias
// =====================================================================
template <int H, int MODE>
__global__ void gat_node_kernel(const int* __restrict__ rowptr, const int* __restrict__ srcs,
                                const float* __restrict__ el, const float* __restrict__ er,
                                const float* __restrict__ z, const float* __restrict__ bias,
                                float* __restrict__ out) {
  int node = (int)((blockIdx.x * blockDim.x + threadIdx.x) >> 5);
  int lane = threadIdx.x & 31;
  if (node >= NNODES) return;                      // wave-uniform guard
  const int base = rowptr[node];
  const int deg = rowptr[node + 1] - base;
  const int* __restrict__ sp = srcs + base;

  float acc0 = 0.f, acc1 = 0.f, acc2 = 0.f, acc3 = 0.f;

  if (H == 8) {
    const int myh = lane & 7;                      // lanes 0-7 own the 8 heads
    const float erh = er[node * 8 + myh];
    float mx = -3.402823466e38f;
    for (int j = 0; j < deg; ++j) {
      int s = sp[j];
      mx = fmaxf(mx, lrelu(el[s * 8 + myh] + erh));
    }
    float dsum = 0.f;
    for (int j = 0; j < deg; ++j) {
      int s = sp[j];
      dsum += expf(lrelu(el[s * 8 + myh] + erh) - mx);
    }
    const float inv = (deg > 0) ? 1.f / dsum : 0.f;
    for (int j = 0; j < deg; ++j) {
      int s = sp[j];
      float a_me = expf(lrelu(el[s * 8 + myh] + erh) - mx) * inv;
      const float* zp = z + (size_t)s * DH;
      float a0 = __shfl(a_me, (lane +  0) >> 4, 32);
      float a1 = __shfl(a_me, (lane + 32) >> 4, 32);
      float a2 = __shfl(a_me, (lane + 64) >> 4, 32);
      float a3 = __shfl(a_me, (lane + 96) >> 4, 32);
      acc0 += a0 * zp[lane];
      acc1 += a1 * zp[lane + 32];
      acc2 += a2 * zp[lane + 64];
      acc3 += a3 * zp[lane + 96];
    }
  } else {  // H == 1: scalar attention, uniform across lanes
    const float erh = er[node];
    float mx = -3.402823466e38f;
    for (int j = 0; j < deg; ++j) mx = fmaxf(mx, lrelu(el[sp[j]] + erh));
    float dsum = 0.f;
    for (int j = 0; j < deg; ++j) dsum += expf(lrelu(el[sp[j]] + erh) - mx);
    const float inv = (deg > 0) ? 1.f / dsum : 0.f;
    for (int j = 0; j < deg; ++j) {
      int s = sp[j];
      float a = expf(lrelu(el[s] + erh) - mx) * inv;
      const float* zp = z + (size_t)s * DH;
      acc0 += a * zp[lane];
      acc1 += a * zp[lane + 32];
      acc2 += a * zp[lane + 64];
      acc3 += a * zp[lane + 96];
    }
  }

  float* op = out + (size_t)node * DH;
  float r0 = acc0 + bias[lane],      r1 = acc1 + bias[lane + 32];
  float r2 = acc2 + bias[lane + 64], r3 = acc3 + bias[lane + 96];
  if (MODE == 1) { r0 = eluf(r0); r1 = eluf(r1); r2 = eluf(r2); r3 = eluf(r3); }
  op[lane] = r0; op[lane + 32] = r1; op[lane + 64] = r2; op[lane + 96] = r3;
}

// =====================================================================
// semantic attention over M=2 metapaths (wave per node)
// =====================================================================
__global__ void semantic_kernel(const float* __restrict__ hs, const float* __restrict__ Wsem,
                                float* __restrict__ hfin) {
  int wave = (int)((blockIdx.x * blockDim.x + threadIdx.x) >> 5);
  int lane = threadIdx.x & 31;
  if (wave >= NNODES) return;
  const float* h0p = hs + (size_t)wave * DH;
  const float* h1p = hs + (size_t)NNODES * DH + (size_t)wave * DH;
  float v0[4], v1[4];
  float s0 = 0.f, s1 = 0.f;
#pragma unroll
  for (int j = 0; j < 4; ++j) {
    int c = j * 32 + lane;
    v0[j] = h0p[c]; v1[j] = h1p[c];
    float wv = Wsem[c];
    s0 += v0[j] * wv; s1 += v1[j] * wv;
  }
#pragma unroll
  for (int m = 16; m > 0; m >>= 1) { s0 += __shfl_xor(s0, m, 32); s1 += __shfl_xor(s1, m, 32); }
  float mx = fmaxf(s0, s1);
  float e0 = expf(s0 - mx), e1 = expf(s1 - mx);
  float inv = 1.f / (e0 + e1);
  float a0 = e0 * inv, a1 = e1 * inv;
  float* fp = hfin + (size_t)wave * DH;
#pragma unroll
  for (int j = 0; j < 4; ++j) {
    int c = j * 32 + lane;
    fp[c] = a0 * v0[j] + a1 * v1[j];
  }
}

// =====================================================================
// host orchestration
// =====================================================================
extern "C" void kernel_launch(void* const* d_in, const int* in_sizes, int n_in,
                              void* d_out, int out_size, void* d_ws, size_t ws_size,
                              hipStream_t stream) {
  const float* feats = (const float*)d_in[0];
  const float* Wp    = (const float*)d_in[1];
  const float* bp    = (const float*)d_in[2];
  const float* W1    = (const float*)d_in[3];
  const float* al1   = (const float*)d_in[4];
  const float* ar1   = (const float*)d_in[5];
  const float* b1    = (const float*)d_in[6];
  const float* W2    = (const float*)d_in[7];
  const float* al2   = (const float*)d_in[8];
  const float* ar2   = (const float*)d_in[9];
  const float* b2    = (const float*)d_in[10];
  const float* Wsem  = (const float*)d_in[11];
  const float* Wcls  = (const float*)d_in[12];
  const float* bcls  = (const float*)d_in[13];
  const int*   src   = (const int*)d_in[14];
  const int*   dst   = (const int*)d_in[15];
  float* out = (float*)d_out;

  const size_t N128 = (size_t)NNODES * DH;
  const size_t N8   = (size_t)NNODES * NHEADS;

  float* ws   = (float*)d_ws;
  float* h0   = ws;                  // [N,128]
  float* zbuf = h0 + N128;           // [N,128]
  float* abuf = zbuf + N128;         // [N,128] h1 / hfin
  float* hs   = abuf + N128;         // [2,N,128]
  float* el   = hs + 2 * N128;       // [N,8]
  float* er   = el + N8;             // [N,8]
  int* ibase  = (int*)(er + N8);
  int* rowptr0 = ibase;              // [N+1]
  int* rowptr1 = rowptr0 + (NNODES + 1);
  int* srcs0   = rowptr1 + (NNODES + 1);  // [E]
  int* srcs1   = srcs0 + NEDGES;          // [E]
  int* indeg   = srcs1 + NEDGES;          // [N]
  int* fill    = indeg + NNODES;          // [N]

  const int THREADS = 256;
  const int edgeBlocks = (NEDGES + THREADS - 1) / THREADS;
  const int nodeWaveBlocks = (int)(((long long)NNODES * 32 + THREADS - 1) / THREADS);
  const int nhBlocks = (NNODES * NHEADS + THREADS - 1) / THREADS;
  const int rowTileBlocks = NNODES / 16;   // 3125
  const int clsBlocks = (int)(((long long)(NNODES / 16) * 32 + THREADS - 1) / THREADS);

  // ---- CSR build (once per metapath) ----
  int* rowptrs[2] = {rowptr0, rowptr1};
  int* srcss[2]   = {srcs0, srcs1};
  for (int m = 0; m < NMETA; ++m) {
    const int* srcm = src + (size_t)m * NEDGES;
    const int* dstm = dst + (size_t)m * NEDGES;
    hipMemsetAsync(indeg, 0, NNODES * sizeof(int), stream);
    csr_count_kernel<<<edgeBlocks, THREADS, 0, stream>>>(dstm, indeg);
    csr_scan_kernel<<<1, 1024, 0, stream>>>(indeg, rowptrs[m], fill);
    csr_fill_kernel<<<edgeBlocks, THREADS, 0, stream>>>(srcm, dstm, fill, srcss[m]);
  }

  // ---- projection: h0 = elu(feats @ Wp + bp) ----
  wmma_gemm_staged_kernel<1, DIN><<<rowTileBlocks, THREADS, 0, stream>>>(feats, Wp, bp, h0);

  for (int m = 0; m < NMETA; ++m) {
    // GAT layer 1 (8 heads x 16)
    wmma_gemm_staged_kernel<0, DH><<<rowTileBlocks, THREADS, 0, stream>>>(
        h0, W1 + (size_t)m * DH * DH, nullptr, zbuf);
    elr8_kernel<<<nhBlocks, THREADS, 0, stream>>>(zbuf, al1 + m * DH, ar1 + m * DH, el, er);
    gat_node_kernel<8, 1><<<nodeWaveBlocks, THREADS, 0, stream>>>(
        rowptrs[m], srcss[m], el, er, zbuf, b1 + m * DH, abuf);

    // GAT layer 2 (1 head x 128)
    wmma_gemm_staged_kernel<0, DH><<<rowTileBlocks, THREADS, 0, stream>>>(
        abuf, W2 + (size_t)m * DH * DH, nullptr, zbuf);
    elr1_kernel<<<nodeWaveBlocks, THREADS, 0, stream>>>(zbuf, al2 + m * DH, ar2 + m * DH, el, er);
    gat_node_kernel<1, 2><<<nodeWaveBlocks, THREADS, 0, stream>>>(
        rowptrs[m], srcss[m], el, er, zbuf, b2 + m * DH, hs + (size_t)m * N128);
  }

  // ---- semantic attention -> abuf ----
  semantic_kernel<<<nodeWaveBlocks, THREADS, 0, stream>>>(hs, Wsem, abuf);

  // ---- classifier: out = hfin @ Wcls + bcls ----
  wmma_gemm_direct_kernel<2><<<clsBlocks, THREADS, 0, stream>>>(
      abuf, Wcls, bcls, out, NNODES, DH, NCLS);
}